// AdvancedBookRecommenderGNN_8340826488853
// MI455X (gfx1250) — compile-verified
//
#include <hip/hip_runtime.h>
#include <math.h>

// ---- problem constants (fixed by the reference) ----
#define NN   60000       // nodes
#define NE   1200000     // edges
#define NET  1260000     // edges + self loops
#define HD   4           // GAT heads
#define CH   32          // channels per head
#define F1   128         // H*C
#define F2   64
#define F3   32
#define BN_EPS 1e-5f

typedef float v2f __attribute__((ext_vector_type(2)));
typedef float v8f __attribute__((ext_vector_type(8)));

// V_WMMA_F32_16X16X4_F32 : D[16x16] = A[16x4] * B[4x16] + C   (fp32, wave32)
__device__ __forceinline__ v8f wmma4(v2f a, v2f b, v8f c) {
  return __builtin_amdgcn_wmma_f32_16x16x4_f32(false, a, false, b, (short)0, c, false, false);
}

__device__ __forceinline__ float elu1(float x) { return x > 0.0f ? x : expm1f(x); }

__device__ __forceinline__ void atomicMaxF(float* addr, float val) {
  unsigned int* ua = (unsigned int*)addr;
  unsigned int cur = *ua;
  while (__uint_as_float(cur) < val) {
    unsigned int prev = atomicCAS(ua, cur, __float_as_uint(val));
    if (prev == cur) break;
    cur = prev;
  }
}

// ---------------- fills ----------------
__global__ void k_fill(float* __restrict__ p, float v, int n) {
  int i = blockIdx.x * blockDim.x + threadIdx.x;
  if (i < n) p[i] = v;
}

// ---------------- GAT: xw = x @ gat_w^T  (N x 64 @ 64 x 128) ----------------
// one wave per 16x16 tile; 3750 M-tiles x 8 N-tiles = 30000 tiles = 3750 blocks of 8 waves
__global__ void k_gemm_xw(const float* __restrict__ x, const float* __restrict__ gw,
                          float* __restrict__ xw) {
  const int lane = threadIdx.x & 31;
  const int tile = blockIdx.x * 8 + (threadIdx.x >> 5);
  const int tm = tile >> 3, tn = tile & 7;
  const int l16 = lane & 15, hf = lane >> 4;
  const int m0 = tm * 16, n0 = tn * 16;
  const float* __restrict__ ar = x  + (size_t)(m0 + l16) * 64;   // A row (node features)
  const float* __restrict__ br = gw + (size_t)(n0 + l16) * 64;   // B col = gat_w row
  v8f acc = {};
  #pragma unroll
  for (int k = 0; k < 64; k += 4) {
    v2f a, b;
    a[0] = ar[k + 2 * hf]; a[1] = ar[k + 2 * hf + 1];
    b[0] = br[k + 2 * hf]; b[1] = br[k + 2 * hf + 1];
    acc = wmma4(a, b, acc);
  }
  #pragma unroll
  for (int j = 0; j < 8; ++j)
    xw[(size_t)(m0 + 8 * hf + j) * F1 + n0 + l16] = acc[j];
}

// a_src[n,h] = <xw[n,h,:], att_src[h,:]> ; same for dst
__global__ void k_att(const float* __restrict__ xw, const float* __restrict__ as,
                      const float* __restrict__ ad, float* __restrict__ asv,
                      float* __restrict__ adv) {
  int t = blockIdx.x * blockDim.x + threadIdx.x;
  if (t >= NN * HD) return;
  int n = t >> 2, h = t & 3;
  const float* row = xw + (size_t)n * F1 + h * CH;
  const float* ws  = as + h * CH;
  const float* wd  = ad + h * CH;
  float s0 = 0.f, s1 = 0.f;
  #pragma unroll
  for (int c = 0; c < CH; ++c) { float v = row[c]; s0 += v * ws[c]; s1 += v * wd[c]; }
  asv[t] = s0; adv[t] = s1;
}

// per-edge leaky-relu logits; segment max via float atomic-max; also SAGE degree
__global__ void k_edge_logits(const int* __restrict__ src, const int* __restrict__ dst,
                              const float* __restrict__ asv, const float* __restrict__ adv,
                              float* __restrict__ lg, float* __restrict__ mx,
                              float* __restrict__ deg) {
  int e = blockIdx.x * blockDim.x + threadIdx.x;
  if (e >= NET) return;
  int s, d;
  if (e < NE) { s = src[e]; d = dst[e]; atomicAdd(&deg[d], 1.0f); }
  else        { s = e - NE; d = s; }
  #pragma unroll
  for (int h = 0; h < HD; ++h) {
    float l = asv[s * 4 + h] + adv[d * 4 + h];
    l = l > 0.f ? l : 0.2f * l;
    lg[(size_t)e * 4 + h] = l;
    atomicMaxF(&mx[d * 4 + h], l);
  }
}

// ex = exp(logit - mx[dst]); accumulate segment sum; overwrite logit buffer with ex
__global__ void k_edge_expsum(const int* __restrict__ src, const int* __restrict__ dst,
                              float* __restrict__ lg, const float* __restrict__ mx,
                              float* __restrict__ ss) {
  int e = blockIdx.x * blockDim.x + threadIdx.x;
  if (e >= NET) return;
  int d = (e < NE) ? dst[e] : e - NE;
  #pragma unroll
  for (int h = 0; h < HD; ++h) {
    float ex = expf(lg[(size_t)e * 4 + h] - mx[d * 4 + h]);
    lg[(size_t)e * 4 + h] = ex;
    atomicAdd(&ss[d * 4 + h], ex);
  }
}

// weighted scatter: acc[dst, c0..c0+3] += alpha * xw[src, c0..c0+3]
// one thread per edge x channel-quad: b128 read of xw, 4 scalar f32 atomics
__global__ void k_gat_scatter(const int* __restrict__ src, const int* __restrict__ dst,
                              const float* __restrict__ xw, const float* __restrict__ ex,
                              const float* __restrict__ ss, float* __restrict__ acc) {
  int t = blockIdx.x * 256 + threadIdx.x;
  if (t >= NET * 32) return;
  int e = t >> 5, q = t & 31;          // q: which 4-channel quad (0..31)
  int c0 = q * 4, h = q >> 3;          // 8 quads per head
  int s, d;
  if (e < NE) { s = src[e]; d = dst[e]; } else { s = e - NE; d = s; }
  float alpha = ex[(size_t)e * 4 + h] / (ss[d * 4 + h] + 1e-16f);
  const float4 xv = *(const float4*)(xw + (size_t)s * F1 + c0);
  float* ap = acc + (size_t)d * F1 + c0;
  atomicAdd(ap + 0, xv.x * alpha);
  atomicAdd(ap + 1, xv.y * alpha);
  atomicAdd(ap + 2, xv.z * alpha);
  atomicAdd(ap + 3, xv.w * alpha);
}

__global__ void k_gat_post(const float* __restrict__ acc, const float* __restrict__ gb,
                           const float* __restrict__ g, const float* __restrict__ b,
                           const float* __restrict__ m, const float* __restrict__ v,
                           float* __restrict__ h1) {
  int t = blockIdx.x * blockDim.x + threadIdx.x;
  if (t >= NN * F1) return;
  int c = t & 127;
  float x = elu1(acc[t] + gb[c]);
  h1[t] = (x - m[c]) * rsqrtf(v[c] + BN_EPS) * g[c] + b[c];
}

// ---------------- SAGE ----------------
// neighbor sum: one thread per edge x channel-quad, b128 read of h1
__global__ void k_sage_gather(const int* __restrict__ src, const int* __restrict__ dst,
                              const float* __restrict__ h1, float* __restrict__ nbr) {
  int t = blockIdx.x * 256 + threadIdx.x;
  if (t >= NE * 32) return;
  int e = t >> 5, c0 = (t & 31) * 4;
  const float4 hv = *(const float4*)(h1 + (size_t)src[e] * F1 + c0);
  float* np = nbr + (size_t)dst[e] * F1 + c0;
  atomicAdd(np + 0, hv.x);
  atomicAdd(np + 1, hv.y);
  atomicAdd(np + 2, hv.z);
  atomicAdd(np + 3, hv.w);
}

// h2 = (nbr/deg) @ wl^T + bl + h1 @ wr^T, then ELU + BN2.  K=128 twice, fused epilogue.
// 3750 M-tiles x 4 N-tiles = 15000 tiles = 1875 blocks of 8 waves
__global__ void k_sage_gemm(const float* __restrict__ nbr, const float* __restrict__ h1,
                            const float* __restrict__ deg,
                            const float* __restrict__ wl, const float* __restrict__ bl,
                            const float* __restrict__ wr,
                            const float* __restrict__ g, const float* __restrict__ bb,
                            const float* __restrict__ bm, const float* __restrict__ bv,
                            float* __restrict__ h2) {
  const int lane = threadIdx.x & 31;
  const int tile = blockIdx.x * 8 + (threadIdx.x >> 5);
  const int tm = tile >> 2, tn = tile & 3;
  const int l16 = lane & 15, hf = lane >> 4;
  const int m0 = tm * 16, n0 = tn * 16;
  const float sc = 1.0f / fmaxf(deg[m0 + l16], 1.0f);   // per-lane row = m0+l16
  const float* a1 = nbr + (size_t)(m0 + l16) * F1;
  const float* a2 = h1  + (size_t)(m0 + l16) * F1;
  const float* b1 = wl  + (size_t)(n0 + l16) * F1;
  const float* b2 = wr  + (size_t)(n0 + l16) * F1;
  v8f acc = {};
  for (int k = 0; k < F1; k += 4) {
    v2f a, b;
    a[0] = a1[k + 2 * hf] * sc; a[1] = a1[k + 2 * hf + 1] * sc;
    b[0] = b1[k + 2 * hf];      b[1] = b1[k + 2 * hf + 1];
    acc = wmma4(a, b, acc);
  }
  for (int k = 0; k < F1; k += 4) {
    v2f a, b;
    a[0] = a2[k + 2 * hf]; a[1] = a2[k + 2 * hf + 1];
    b[0] = b2[k + 2 * hf]; b[1] = b2[k + 2 * hf + 1];
    acc = wmma4(a, b, acc);
  }
  const int col = n0 + l16;
  const float gc = g[col], bc = bb[col], mc = bm[col],
              rc = rsqrtf(bv[col] + BN_EPS), blc = bl[col];
  #pragma unroll
  for (int j = 0; j < 8; ++j) {
    int row = m0 + 8 * hf + j;
    float x = elu1(acc[j] + blc);
    h2[(size_t)row * F2 + col] = (x - mc) * rc * gc + bc;
  }
}

// ---------------- GCN ----------------
// hw = h2 @ gcn_w^T  (N x 64 @ 64 x 32).  3750 x 2 tiles = 7500; 938 blocks (last 4 waves idle)
__global__ void k_gcn_gemm(const float* __restrict__ h2, const float* __restrict__ w,
                           float* __restrict__ hw) {
  const int lane = threadIdx.x & 31;
  const int tile = blockIdx.x * 8 + (threadIdx.x >> 5);
  if (tile >= 7500) return;                              // wave-uniform guard
  const int tm = tile >> 1, tn = tile & 1;
  const int l16 = lane & 15, hf = lane >> 4;
  const int m0 = tm * 16, n0 = tn * 16;
  const float* ar = h2 + (size_t)(m0 + l16) * F2;
  const float* br = w  + (size_t)(n0 + l16) * F2;
  v8f acc = {};
  #pragma unroll
  for (int k = 0; k < F2; k += 4) {
    v2f a, b;
    a[0] = ar[k + 2 * hf]; a[1] = ar[k + 2 * hf + 1];
    b[0] = br[k + 2 * hf]; b[1] = br[k + 2 * hf + 1];
    acc = wmma4(a, b, acc);
  }
  #pragma unroll
  for (int j = 0; j < 8; ++j)
    hw[(size_t)(m0 + 8 * hf + j) * F3 + n0 + l16] = acc[j];
}

// symmetric-norm scatter with self loops: deg_sl = deg + 1
// one thread per edge x channel-quad (8 quads per 32-ch row)
__global__ void k_gcn_scatter(const int* __restrict__ src, const int* __restrict__ dst,
                              const float* __restrict__ hw, const float* __restrict__ deg,
                              float* __restrict__ acc3) {
  int t = blockIdx.x * 256 + threadIdx.x;
  if (t >= NET * 8) return;
  int e = t >> 3, c0 = (t & 7) * 4;
  int s, d;
  if (e < NE) { s = src[e]; d = dst[e]; } else { s = e - NE; d = s; }
  float nrm = rsqrtf(deg[s] + 1.0f) * rsqrtf(deg[d] + 1.0f);
  const float4 hv = *(const float4*)(hw + (size_t)s * F3 + c0);
  float* ap = acc3 + (size_t)d * F3 + c0;
  atomicAdd(ap + 0, hv.x * nrm);
  atomicAdd(ap + 1, hv.y * nrm);
  atomicAdd(ap + 2, hv.z * nrm);
  atomicAdd(ap + 3, hv.w * nrm);
}

__global__ void k_gcn_post(const float* __restrict__ acc3, const float* __restrict__ gb,
                           const float* __restrict__ g, const float* __restrict__ b,
                           const float* __restrict__ m, const float* __restrict__ v,
                           float* __restrict__ out) {
  int t = blockIdx.x * blockDim.x + threadIdx.x;
  if (t >= NN * F3) return;
  int c = t & 31;
  float x = elu1(acc3[t] + gb[c]);
  out[t] = (x - m[c]) * rsqrtf(v[c] + BN_EPS) * g[c] + b[c];
}

extern "C" void kernel_launch(void* const* d_in, const int* in_sizes, int n_in,
                              void* d_out, int out_size, void* d_ws, size_t ws_size,
                              hipStream_t stream) {
  const float* x     = (const float*)d_in[0];
  const int*   ei    = (const int*)d_in[1];
  const int*   src   = ei;          // edge_index row 0
  const int*   dst   = ei + NE;     // edge_index row 1
  const float* gat_w = (const float*)d_in[2];
  const float* att_s = (const float*)d_in[3];
  const float* att_d = (const float*)d_in[4];
  const float* gat_b = (const float*)d_in[5];
  const float* bn1g  = (const float*)d_in[6];
  const float* bn1b  = (const float*)d_in[7];
  const float* bn1m  = (const float*)d_in[8];
  const float* bn1v  = (const float*)d_in[9];
  const float* swl   = (const float*)d_in[10];
  const float* sbl   = (const float*)d_in[11];
  const float* swr   = (const float*)d_in[12];
  const float* bn2g  = (const float*)d_in[13];
  const float* bn2b  = (const float*)d_in[14];
  const float* bn2m  = (const float*)d_in[15];
  const float* bn2v  = (const float*)d_in[16];
  const float* gcw   = (const float*)d_in[17];
  const float* gcb   = (const float*)d_in[18];
  const float* bn3g  = (const float*)d_in[19];
  const float* bn3b  = (const float*)d_in[20];
  const float* bn3m  = (const float*)d_in[21];
  const float* bn3v  = (const float*)d_in[22];

  float* ws = (float*)d_ws;
  size_t o = 0;
  float* XW  = ws + o; o += (size_t)NN * F1;   // x @ gat_w^T
  float* LG  = ws + o; o += (size_t)NET * HD;  // logits -> exp numerators
  float* ASV = ws + o; o += (size_t)NN * HD;
  float* ADV = ws + o; o += (size_t)NN * HD;
  float* MX  = ws + o; o += (size_t)NN * HD;
  float* SS  = ws + o; o += (size_t)NN * HD;
  float* AC1 = ws + o; o += (size_t)NN * F1;   // GAT accumulator
  float* H1  = ws + o; o += (size_t)NN * F1;
  float* DEG = ws + o; o += (size_t)NN;
  // aliases (stream-ordered reuse):
  float* NBR = XW;                    // neighbor sums (zeroed after GAT scatter)
  float* H2  = AC1;                   // SAGE output overwrites GAT accumulator
  float* HW  = AC1 + (size_t)NN * F2; // GCN pre-agg lives in the rest of AC1
  float* AC3 = LG;                    // GCN accumulator (zeroed after exp buffer use)
  float* OUT = (float*)d_out;

  const int B = 256;
  // ---- init accumulators (every call: graph replay must be self-contained) ----
  k_fill<<<(NN * HD + B - 1) / B, B, 0, stream>>>(MX, -1e30f, NN * HD);
  k_fill<<<(NN * HD + B - 1) / B, B, 0, stream>>>(SS, 0.f, NN * HD);
  k_fill<<<(NN * F1 + B - 1) / B, B, 0, stream>>>(AC1, 0.f, NN * F1);
  k_fill<<<(NN + B - 1) / B, B, 0, stream>>>(DEG, 0.f, NN);

  // ---- GAT ----
  k_gemm_xw<<<3750, 256, 0, stream>>>(x, gat_w, XW);
  k_att<<<(NN * HD + B - 1) / B, B, 0, stream>>>(XW, att_s, att_d, ASV, ADV);
  k_edge_logits<<<(NET + B - 1) / B, B, 0, stream>>>(src, dst, ASV, ADV, LG, MX, DEG);
  k_edge_expsum<<<(NET + B - 1) / B, B, 0, stream>>>(src, dst, LG, MX, SS);
  k_gat_scatter<<<(NET * 32) / B, B, 0, stream>>>(src, dst, XW, LG, SS, AC1);
  k_gat_post<<<(NN * F1 + B - 1) / B, B, 0, stream>>>(AC1, gat_b, bn1g, bn1b, bn1m, bn1v, H1);

  // ---- zero aliased accumulators now that their old contents are consumed ----
  k_fill<<<(NN * F1 + B - 1) / B, B, 0, stream>>>(NBR, 0.f, NN * F1);
  k_fill<<<(NN * F3 + B - 1) / B, B, 0, stream>>>(AC3, 0.f, NN * F3);

  // ---- SAGE ----
  k_sage_gather<<<(NE * 32) / B, B, 0, stream>>>(src, dst, H1, NBR);
  k_sage_gemm<<<1875, 256, 0, stream>>>(NBR, H1, DEG, swl, sbl, swr,
                                        bn2g, bn2b, bn2m, bn2v, H2);

  // ---- GCN ----
  k_gcn_gemm<<<938, 256, 0, stream>>>(H2, gcw, HW);
  k_gcn_scatter<<<(NET * 8 + B - 1) / B, B, 0, stream>>>(src, dst, HW, DEG, AC3);
  k_gcn_post<<<(NN * F3 + B - 1) / B, B, 0, stream>>>(AC3, gcb, bn3g, bn3b, bn3m, bn3v, OUT);
}